// TemporalUnfold1d_19490561589739
// MI455X (gfx1250) — compile-verified
//
#include <hip/hip_runtime.h>

// TemporalUnfold1d: out[b, kk*C + c, t] = x[b, c, t + kk - PAD], zero-padded left.
// Fixed shape: B=16, C=256, T=8192, K=4, D=1, PAD=3. f32 in/out.
//
// Pure bandwidth kernel: 128 MiB read + 512 MiB write = 640 MiB -> ~28.8 us at
// 23.3 TB/s HBM. Strategy: LDS-stage each 2048-float row tile (+3 halo, skewed
// by PAD so shifted gathers are branch-free), read global exactly once with
// aligned NT b128 loads, emit only aligned NT b128 stores. 3D grid kills all
// div/mod in the preamble.

typedef float v4f __attribute__((ext_vector_type(4)));

constexpr int KSZ      = 4;      // kernel size
constexpr int PAD      = 3;      // (K-1)*D causal left pad
constexpr int B        = 16;
constexpr int C        = 256;
constexpr int T        = 8192;
constexpr int TILE     = 2048;   // floats of one row handled per block
constexpr int NTHREADS = 256;    // 8 waves (wave32)
constexpr int VEC_IT   = TILE / (4 * NTHREADS);  // float4s per thread = 2
constexpr int TILES_PER_ROW = T / TILE;          // 4

__global__ __launch_bounds__(NTHREADS)
void temporal_unfold1d_kernel(const float* __restrict__ x,
                              float* __restrict__ out)
{
    // +PAD skew at the front so lds[PAD + j] = xrow[t0 + j]
    __shared__ float lds[TILE + PAD];

    const int tile = blockIdx.x;        // 0..3   tile within row
    const int c    = blockIdx.y;        // 0..255 channel
    const int b    = blockIdx.z;        // 0..15  batch
    const int t0   = tile * TILE;
    const int tid  = threadIdx.x;

    const float* __restrict__ xrow = x + ((long)b * C + c) * T;

    // ---- Phase 1: stage tile into LDS (aligned NT b128 global loads) ----
    const v4f* __restrict__ src4 =
        reinterpret_cast<const v4f*>(xrow + t0);   // 16B aligned (t0 % 2048 == 0)
#pragma unroll
    for (int it = 0; it < VEC_IT; ++it) {
        const int idx = it * NTHREADS + tid;       // float4 index within tile
        v4f v = __builtin_nontemporal_load(src4 + idx);
        const int base = PAD + idx * 4;            // skewed word stores into LDS
        lds[base + 0] = v.x;
        lds[base + 1] = v.y;
        lds[base + 2] = v.z;
        lds[base + 3] = v.w;
    }
    // 3-element left halo: xrow[t0-3 .. t0-1], zero at the row start
    if (tid < PAD) {
        const int g = t0 - PAD + tid;
        lds[tid] = (g >= 0) ? xrow[g] : 0.0f;
    }
    __syncthreads();

    // ---- Phase 2: 4 shifted copies, each as aligned NT b128 stores ----
#pragma unroll
    for (int kk = 0; kk < KSZ; ++kk) {
        v4f* __restrict__ dst4 = reinterpret_cast<v4f*>(
            out + (((long)b * KSZ + kk) * C + c) * T + t0);
#pragma unroll
        for (int it = 0; it < VEC_IT; ++it) {
            const int idx = it * NTHREADS + tid;
            const int m   = idx * 4 + kk;          // lds[m + e] = x[t + kk - PAD]
            v4f w;
            w.x = lds[m + 0];
            w.y = lds[m + 1];
            w.z = lds[m + 2];
            w.w = lds[m + 3];
            __builtin_nontemporal_store(w, dst4 + idx);
        }
    }
}

extern "C" void kernel_launch(void* const* d_in, const int* in_sizes, int n_in,
                              void* d_out, int out_size, void* d_ws, size_t ws_size,
                              hipStream_t stream)
{
    (void)in_sizes; (void)n_in; (void)d_ws; (void)ws_size; (void)out_size;

    const float* x  = (const float*)d_in[0];
    float*      out = (float*)d_out;

    dim3 grid(TILES_PER_ROW, C, B);     // 4 x 256 x 16 = 16384 blocks
    temporal_unfold1d_kernel<<<grid, NTHREADS, 0, stream>>>(x, out);
}